// CEM_22754736734575
// MI455X (gfx1250) — compile-verified
//
#include <hip/hip_runtime.h>
#include <math.h>
#include <stdint.h>

// ---------------------------------------------------------------------------
// CEM forward on gfx1250 (MI455X), fp32 WMMA path (V_WMMA_F32_16X16X4_F32).
// Memory-bound: x = 512MB streamed once (coalesced, async-to-LDS); weights
// (~12MB) live in the 192MB L2 across batch tiles. One wave owns one concept
// end-to-end; all intermediates stay in registers / per-wave LDS.
// ---------------------------------------------------------------------------

#define BDIM 8192
#define FDIM 256
#define CDIM 64
#define HDIM 64
#define EDIM 16
#define FC   16                         // f-rows staged per chunk

typedef float v2f __attribute__((ext_vector_type(2)));
typedef float v8f __attribute__((ext_vector_type(8)));

#define XS_CP 9                         // padded c-dim of x stage tile (conflict-free A reads)
#define BUF_S 68                        // padded row stride of 16x64 hidden buffer
#define CMB_S 36                        // padded row stride of 16x32 comb buffer

#define WV_FLOATS (16 * BUF_S + 16 * CMB_S + 16)    // 1680 floats / wave
#define XS_FLOATS (FC * 16 * XS_CP)                 // 2304 floats shared
// total: 2304 + 8*1680 = 15744 floats = 62976 B  (< 64KB, no attribute needed)

#if defined(__has_builtin)
#if __has_builtin(__builtin_amdgcn_global_load_async_to_lds_b32) && \
    __has_builtin(__builtin_amdgcn_s_wait_asynccnt)
#define USE_ASYNC_STAGE 1
#endif
#endif
#ifndef USE_ASYNC_STAGE
#define USE_ASYNC_STAGE 0
#endif

typedef __attribute__((address_space(1))) int gint;   // global (AS1) int
typedef __attribute__((address_space(3))) int lint;   // LDS (AS3) int

__device__ __forceinline__ v8f wmma_f32(v2f a, v2f b, v8f c) {
  // D = A(16x4) * B(4x16) + C, fp32, wave32
  return __builtin_amdgcn_wmma_f32_16x16x4_f32(false, a, false, b, (short)0, c,
                                               false, false);
}

// Store a 16x16 D tile (lane: n = lane&15, rows mb..mb+7) into padded LDS buf.
__device__ __forceinline__ void store_d(float* buf, int os, int n0, int n, int mb,
                                        v8f d, bool dorelu) {
#pragma unroll
  for (int r = 0; r < 8; ++r) {
    float v = d[r];
    if (dorelu) v = fmaxf(v, 0.0f);
    buf[(mb + r) * os + n0 + n] = v;
  }
}

// relu(A[16xK] @ W[Kx64] + bias) -> obuf (A from padded LDS abuf; in-place ok:
// all A reads complete before the D store).
__device__ __forceinline__ void layer_to64(const float* abuf, int as, int K,
                                           const float* __restrict__ W,
                                           const float* __restrict__ bias,
                                           float* obuf, int os, bool dorelu,
                                           int m, int n, int kb2, int mb) {
  v8f acc[4];
#pragma unroll
  for (int nt = 0; nt < 4; ++nt) {
    float bb = bias[nt * 16 + n];
    v8f a = {bb, bb, bb, bb, bb, bb, bb, bb};
    acc[nt] = a;
  }
  for (int kt = 0; kt < K; kt += 4) {
    int kb = kt + kb2;
    v2f a;
    a.x = abuf[m * as + kb];
    a.y = abuf[m * as + kb + 1];
#pragma unroll
    for (int nt = 0; nt < 4; ++nt) {
      v2f b;
      b.x = W[(size_t)kb * HDIM + nt * 16 + n];
      b.y = W[(size_t)(kb + 1) * HDIM + nt * 16 + n];
      acc[nt] = wmma_f32(a, b, acc[nt]);
    }
  }
#pragma unroll
  for (int nt = 0; nt < 4; ++nt) store_d(obuf, os, nt * 16, n, mb, acc[nt], dorelu);
}

// A[16xK] @ W[Kx16] + bias  (result kept in registers, D layout)
__device__ __forceinline__ v8f layer_to16(const float* abuf, int as, int K,
                                          const float* __restrict__ W,
                                          const float* __restrict__ bias,
                                          int m, int n, int kb2) {
  float bb = bias[n];
  v8f acc = {bb, bb, bb, bb, bb, bb, bb, bb};
  for (int kt = 0; kt < K; kt += 4) {
    int kb = kt + kb2;
    v2f a;
    a.x = abuf[m * as + kb];
    a.y = abuf[m * as + kb + 1];
    v2f b;
    b.x = W[kb * EDIM + n];
    b.y = W[(kb + 1) * EDIM + n];
    acc = wmma_f32(a, b, acc);
  }
  return acc;
}

extern "C" __global__ __launch_bounds__(256) void cem_fused_kernel(
    const float* __restrict__ x,
    const float* __restrict__ pW1, const float* __restrict__ pb1,
    const float* __restrict__ pW2, const float* __restrict__ pb2,
    const float* __restrict__ pW3, const float* __restrict__ pb3,
    const float* __restrict__ nW1, const float* __restrict__ nb1,
    const float* __restrict__ nW2, const float* __restrict__ nb2,
    const float* __restrict__ nW3, const float* __restrict__ nb3,
    const float* __restrict__ rW1, const float* __restrict__ rb1,
    const float* __restrict__ rW2, const float* __restrict__ rb2,
    const float* __restrict__ rW3, const float* __restrict__ rb3,
    float* __restrict__ out_final, float* __restrict__ out_pred) {
  extern __shared__ float lds[];
  float* xs = lds;  // x stage tile [16f][16b][9c-pad]; later reused as fin stage

  const int t = threadIdx.x;
  const int wv = t >> 5;          // wave id = local concept
  const int lane = t & 31;
  const int m = lane & 15;        // row within 16 (A/D)
  const int n = m;                // col within 16 (B/D)
  const int hi = lane >> 4;       // lane half
  const int kb2 = hi * 2;         // K sub-base within a K=4 tile
  const int mb = hi * 8;          // D row base

  float* wbase = lds + XS_FLOATS + wv * WV_FLOATS;
  float* bufH = wbase;                 // 16 x BUF_S (h1/h2/r1/r2, reused)
  float* bufC = bufH + 16 * BUF_S;     // 16 x CMB_S (comb = [pos|neg])
  float* sigb = bufC + 16 * CMB_S;     // 16 sigmoid weights

  const int b0 = blockIdx.y * 16;          // batch tile
  const int c0 = blockIdx.x * 8;           // concept group
  const int c = c0 + wv;

  // ---------------- layer 1 (K = 256, both nets), x staged via LDS ----------
  v8f accP[4], accN[4];
#pragma unroll
  for (int nt = 0; nt < 4; ++nt) {
    float bp = pb1[c * HDIM + nt * 16 + n];
    float bn = nb1[c * HDIM + nt * 16 + n];
    v8f vp = {bp, bp, bp, bp, bp, bp, bp, bp};
    v8f vn = {bn, bn, bn, bn, bn, bn, bn, bn};
    accP[nt] = vp;
    accN[nt] = vn;
  }
  const float* Wp1 = pW1 + (size_t)c * FDIM * HDIM;
  const float* Wn1 = nW1 + (size_t)c * FDIM * HDIM;

  for (int fc = 0; fc < FDIM; fc += FC) {
    __syncthreads();
#if USE_ASYNC_STAGE
    // ASYNCcnt path: x[b0:+16, fc:+16, c0:+8] -> xs[f][b][c] without VGPR hop
#pragma unroll
    for (int it = 0; it < 8; ++it) {
      int idx = it * 256 + t;
      int cc = idx & 7;
      int bb = (idx >> 3) & 15;
      int ff = idx >> 7;  // 0..15
      const float* src =
          x + ((size_t)(b0 + bb) * FDIM + fc + ff) * CDIM + c0 + cc;
      float* dst = xs + (ff * 16 + bb) * XS_CP + cc;
      __builtin_amdgcn_global_load_async_to_lds_b32(
          (gint*)(uintptr_t)src, (lint*)(uint32_t)(uintptr_t)dst, 0, 0);
    }
    __builtin_amdgcn_s_wait_asynccnt(0);
#else
#pragma unroll
    for (int it = 0; it < 4; ++it) {
      int idx = it * 256 + t;
      int cc2 = (idx & 3) * 2;
      int bb = (idx >> 2) & 15;
      int ff = idx >> 6;  // 0..15
      const float* src =
          x + ((size_t)(b0 + bb) * FDIM + fc + ff) * CDIM + c0 + cc2;
      float2 v = *reinterpret_cast<const float2*>(src);
      float* dst = xs + (ff * 16 + bb) * XS_CP + cc2;
      dst[0] = v.x;
      dst[1] = v.y;
    }
#endif
    __syncthreads();
#pragma unroll
    for (int kt = 0; kt < FC / 4; ++kt) {
      int fl = kt * 4 + kb2;  // local K index pair base
      v2f a;
      a.x = xs[(fl * 16 + m) * XS_CP + wv];
      a.y = xs[((fl + 1) * 16 + m) * XS_CP + wv];
      int kg = fc + fl;       // global f
#pragma unroll
      for (int nt = 0; nt < 4; ++nt) {
        v2f bp, bn;
        bp.x = Wp1[(size_t)kg * HDIM + nt * 16 + n];
        bp.y = Wp1[(size_t)(kg + 1) * HDIM + nt * 16 + n];
        accP[nt] = wmma_f32(a, bp, accP[nt]);
        bn.x = Wn1[(size_t)kg * HDIM + nt * 16 + n];
        bn.y = Wn1[(size_t)(kg + 1) * HDIM + nt * 16 + n];
        accN[nt] = wmma_f32(a, bn, accN[nt]);
      }
    }
  }

  // ---------------- p-net layers 2,3 (accN stays live in registers) --------
#pragma unroll
  for (int nt = 0; nt < 4; ++nt) store_d(bufH, BUF_S, nt * 16, n, mb, accP[nt], true);
  layer_to64(bufH, BUF_S, HDIM, pW2 + (size_t)c * HDIM * HDIM, pb2 + c * HDIM,
             bufH, BUF_S, true, m, n, kb2, mb);
  v8f embP = layer_to16(bufH, BUF_S, HDIM, pW3 + (size_t)c * HDIM * EDIM,
                        pb3 + c * EDIM, m, n, kb2);
  store_d(bufC, CMB_S, 0, n, mb, embP, false);    // comb[:,0:16] = pos

  // ---------------- n-net layers 2,3 ---------------------------------------
#pragma unroll
  for (int nt = 0; nt < 4; ++nt) store_d(bufH, BUF_S, nt * 16, n, mb, accN[nt], true);
  layer_to64(bufH, BUF_S, HDIM, nW2 + (size_t)c * HDIM * HDIM, nb2 + c * HDIM,
             bufH, BUF_S, true, m, n, kb2, mb);
  v8f embN = layer_to16(bufH, BUF_S, HDIM, nW3 + (size_t)c * HDIM * EDIM,
                        nb3 + c * EDIM, m, n, kb2);
  store_d(bufC, CMB_S, 16, n, mb, embN, false);   // comb[:,16:32] = neg

  // ---------------- regression net -----------------------------------------
  layer_to64(bufC, CMB_S, 2 * EDIM, rW1 + (size_t)c * 2 * EDIM * HDIM,
             rb1 + c * HDIM, bufH, BUF_S, true, m, n, kb2, mb);   // r1
  layer_to64(bufH, BUF_S, HDIM, rW2 + (size_t)c * HDIM * HDIM, rb2 + c * HDIM,
             bufH, BUF_S, true, m, n, kb2, mb);                   // r2 (in place)

  // pred = r2 @ rW3 (64->1) + rb3 : 64-FMA dot per row on lanes 0..15
  __builtin_amdgcn_wave_barrier();
  if (lane < 16) {
    float s = rb3[c];
    const float* w3 = rW3 + c * HDIM;
#pragma unroll 8
    for (int k = 0; k < HDIM; ++k) s = fmaf(bufH[lane * BUF_S + k], w3[k], s);
    sigb[lane] = 1.0f / (1.0f + __expf(-s));
    out_pred[(size_t)(b0 + lane) * CDIM + c] = s;  // predicted (pre-sigmoid)
  }
  __builtin_amdgcn_wave_barrier();

  // ---------------- blend + transposed output ------------------------------
  float fvals[8];
#pragma unroll
  for (int r = 0; r < 8; ++r) {
    float wgt = sigb[mb + r];
    fvals[r] = embP[r] * wgt + embN[r] * (1.0f - wgt);
  }

  __syncthreads();  // all waves past the x stage; reuse xs as fin stage
  // fin[b][e][c-pad9]  (b = mb+r, e = n, c = wv)
#pragma unroll
  for (int r = 0; r < 8; ++r) xs[((mb + r) * 16 + n) * XS_CP + wv] = fvals[r];
  __syncthreads();

  // coalesced store: out_final[b, e*64 + c], float2 over c
#pragma unroll
  for (int it = 0; it < 4; ++it) {
    int idx = it * 256 + t;
    int cc2 = (idx & 3) * 2;
    int e = (idx >> 2) & 15;
    int bb = idx >> 6;
    float2 v;
    v.x = xs[(bb * 16 + e) * XS_CP + cc2];
    v.y = xs[(bb * 16 + e) * XS_CP + cc2 + 1];
    *reinterpret_cast<float2*>(out_final + (size_t)(b0 + bb) * (EDIM * CDIM) +
                               e * CDIM + c0 + cc2) = v;
  }
}

extern "C" void kernel_launch(void* const* d_in, const int* in_sizes, int n_in,
                              void* d_out, int out_size, void* d_ws,
                              size_t ws_size, hipStream_t stream) {
  const float* x = (const float*)d_in[0];
  const float* pW1 = (const float*)d_in[1];
  const float* pb1 = (const float*)d_in[2];
  const float* pW2 = (const float*)d_in[3];
  const float* pb2 = (const float*)d_in[4];
  const float* pW3 = (const float*)d_in[5];
  const float* pb3 = (const float*)d_in[6];
  const float* nW1 = (const float*)d_in[7];
  const float* nb1 = (const float*)d_in[8];
  const float* nW2 = (const float*)d_in[9];
  const float* nb2 = (const float*)d_in[10];
  const float* nW3 = (const float*)d_in[11];
  const float* nb3 = (const float*)d_in[12];
  const float* rW1 = (const float*)d_in[13];
  const float* rb1 = (const float*)d_in[14];
  const float* rW2 = (const float*)d_in[15];
  const float* rb2 = (const float*)d_in[16];
  const float* rW3 = (const float*)d_in[17];
  const float* rb3 = (const float*)d_in[18];

  float* out_final = (float*)d_out;
  float* out_pred = out_final + (size_t)BDIM * EDIM * CDIM;

  size_t shmem = (size_t)(XS_FLOATS + 8 * WV_FLOATS) * sizeof(float);  // 62976 B

  dim3 grid(CDIM / 8, BDIM / 16);  // c-groups fastest -> L2 reuse of x lines
  cem_fused_kernel<<<grid, 256, shmem, stream>>>(
      x, pW1, pb1, pW2, pb2, pW3, pb3, nW1, nb1, nW2, nb2, nW3, nb3, rW1, rb1,
      rW2, rb2, rW3, rb3, out_final, out_pred);
}